// Model_79388175499475
// MI455X (gfx1250) — compile-verified
//
#include <hip/hip_runtime.h>
#include <math.h>
#include <string.h>
#include <stddef.h>

// ---------------------------------------------------------------------------
// FiLM forecaster restructured as dense linear algebra for gfx1250 (MI455X).
//   out[b,p,j] = mlp_b + sum_t Q[p,t] * xn[b,t,j]
//   Q folds:  per-scale M_k[o,s] (WMMA f32 GEMM, K=8192) -> E_k -> mlp_w
//   M_k[o,s] = sum_{i,m} wr[i,o,m]*Re(Gc_k[i,m,s]) - wi[i,o,m]*Im(Gc_k[i,m,s])
//   Gc_k[i,m,s] = r_m * e^{-i th m s} * T_m(s),  T backward recurrence over phi_d
// Constants (Ad, Bd, phi, E, twiddles) computed on host in f64 per call.
// Gc is stored TRANSPOSED: GcT[s][kappa] so both GEMM panels fill with
// contiguous float4s and both A/B fragments are single 8-byte LDS loads.
// ---------------------------------------------------------------------------

typedef __attribute__((ext_vector_type(2))) float v2f;
typedef __attribute__((ext_vector_type(8))) float v8f;

#define NH     256     // HiPPO state
#define NMODES 32
#define SEQ    512
#define PRED   128
#define CH     128
#define NBATCH 4
#define KTOT   8192    // NH * NMODES
#define KSLICE 1024    // split-K slice (8 slices per matrix)
#define NSTEPS (KSLICE / 32)
#define LPAD   36      // LDS row pad (odd mod-64 spread, 16B aligned rows)

struct HostConsts {
  float phi[(128 + 256 + 512) * NH];      // per scale: [d][i]
  float weR[(128 + 256 + 512) * NMODES];  // per scale: [d][m] =  cos(2pi m d/L)
  float weI[(128 + 256 + 512) * NMODES];  // per scale: [d][m] = -sin(2pi m d/L)
  float rR[3 * NMODES];
  float rI[3 * NMODES];
  float E[3 * PRED * NH];                 // per scale: [p][n]
};

static HostConsts g_hc;  // persistent source for captured H2D memcpy; fully
                         // rewritten (identically) on every call.

// ------------------------------- host constants ----------------------------
static void build_constants() {
  static double A[NH][NH], Bv[NH];
  static double AUG[NH][2 * NH];
  static double IpA[NH][NH], Ad[NH][NH], Minv[NH][NH];
  static double Bd[NH], vcur[NH], vnext[NH];
  const double PI = 3.14159265358979323846;

  for (int i = 0; i < NH; ++i) {
    for (int j = 0; j < NH; ++j) {
      double v = (i < j) ? -1.0 : (((i - j + 1) & 1) ? -1.0 : 1.0);
      A[i][j] = v * (2.0 * i + 1.0);
    }
    Bv[i] = ((i & 1) ? -1.0 : 1.0) * (2.0 * i + 1.0);
  }

  int phiOff = 0, weOff = 0;
  for (int k = 0; k < 3; ++k) {
    const int L = PRED << k;
    const double dt = 1.0 / (double)L;
    const double h = 0.5 * dt;

    // AUG = [I - hA | I]; Gauss-Jordan with partial pivoting (f64).
    for (int i = 0; i < NH; ++i) {
      for (int j = 0; j < NH; ++j) {
        AUG[i][j] = ((i == j) ? 1.0 : 0.0) - h * A[i][j];
        AUG[i][NH + j] = (i == j) ? 1.0 : 0.0;
      }
    }
    for (int c = 0; c < NH; ++c) {
      int piv = c; double best = fabs(AUG[c][c]);
      for (int r = c + 1; r < NH; ++r)
        if (fabs(AUG[r][c]) > best) { best = fabs(AUG[r][c]); piv = r; }
      if (piv != c)
        for (int q = 0; q < 2 * NH; ++q) { double t = AUG[c][q]; AUG[c][q] = AUG[piv][q]; AUG[piv][q] = t; }
      double inv = 1.0 / AUG[c][c];
      for (int q = 0; q < 2 * NH; ++q) AUG[c][q] *= inv;
      for (int r = 0; r < NH; ++r) {
        if (r == c) continue;
        double f = AUG[r][c];
        if (f != 0.0)
          for (int q = 0; q < 2 * NH; ++q) AUG[r][q] -= f * AUG[c][q];
      }
    }
    for (int i = 0; i < NH; ++i)
      for (int j = 0; j < NH; ++j) Minv[i][j] = AUG[i][NH + j];

    for (int i = 0; i < NH; ++i)
      for (int j = 0; j < NH; ++j)
        IpA[i][j] = ((i == j) ? 1.0 : 0.0) + h * A[i][j];
    for (int i = 0; i < NH; ++i)
      for (int j = 0; j < NH; ++j) {
        double a = 0.0;
        for (int t = 0; t < NH; ++t) a += Minv[i][t] * IpA[t][j];
        Ad[i][j] = a;
      }
    for (int i = 0; i < NH; ++i) {
      double a = 0.0;
      for (int t = 0; t < NH; ++t) a += Minv[i][t] * (dt * Bv[t]);
      Bd[i] = a;
    }

    // phi_d = Ad^d Bd
    for (int i = 0; i < NH; ++i) vcur[i] = Bd[i];
    for (int d = 0; d < L; ++d) {
      for (int i = 0; i < NH; ++i) g_hc.phi[phiOff + d * NH + i] = (float)vcur[i];
      if (d + 1 < L) {
        for (int i = 0; i < NH; ++i) {
          double a = 0.0;
          for (int t = 0; t < NH; ++t) a += Ad[i][t] * vcur[t];
          vnext[i] = a;
        }
        memcpy(vcur, vnext, sizeof(vcur));
      }
    }

    // twiddles [d][m]
    for (int d = 0; d < L; ++d)
      for (int m = 0; m < NMODES; ++m) {
        double ang = 2.0 * PI * (double)m * (double)d / (double)L;
        g_hc.weR[weOff + d * NMODES + m] = (float)cos(ang);
        g_hc.weI[weOff + d * NMODES + m] = (float)(-sin(ang));
      }
    // irfft single-sample coefficients at n0 = PRED-1
    for (int m = 0; m < NMODES; ++m) {
      double ang = 2.0 * PI * (double)m * (double)(PRED - 1) / (double)L;
      double c = ((m == 0) ? 1.0 : 2.0) / (double)L;
      g_hc.rR[k * NMODES + m] = (float)(c * cos(ang));
      g_hc.rI[k * NMODES + m] = (float)(c * sin(ang));
    }
    // Legendre readout E_k[p][n] = P_n(1 - 2*(L-128+p)/L)
    for (int p = 0; p < PRED; ++p) {
      double x = 1.0 - 2.0 * (double)(L - PRED + p) / (double)L;
      double p0 = 1.0, p1 = x;
      float* Ep = &g_hc.E[k * PRED * NH + p * NH];
      Ep[0] = 1.0f; Ep[1] = (float)x;
      for (int n = 1; n < NH - 1; ++n) {
        double p2 = ((2.0 * n + 1.0) * x * p1 - (double)n * p0) / (double)(n + 1);
        Ep[n + 1] = (float)p2;
        p0 = p1; p1 = p2;
      }
    }
    phiOff += L * NH;
    weOff  += L * NMODES;
  }
}

// ------------------------------- device kernels ----------------------------

// RevIN: one wave per (b, c)
__global__ void film_revin(const float* __restrict__ x, const float* __restrict__ aw,
                           const float* __restrict__ ab, float* __restrict__ xn) {
  int wave = (blockIdx.x * blockDim.x + threadIdx.x) >> 5;  // 0..511
  int lane = threadIdx.x & 31;
  int b = wave >> 7, c = wave & 127;
  const float* xp = x + (size_t)b * SEQ * CH + c;
  float v[16], s = 0.f, s2 = 0.f;
#pragma unroll
  for (int q = 0; q < 16; ++q) {
    float t = xp[(size_t)(lane + q * 32) * CH];
    v[q] = t; s += t; s2 += t * t;
  }
#pragma unroll
  for (int off = 16; off; off >>= 1) {
    s  += __shfl_xor(s, off, 32);
    s2 += __shfl_xor(s2, off, 32);
  }
  float mean = s * (1.f / SEQ);
  float var  = s2 * (1.f / SEQ) - mean * mean;
  float inv  = 1.f / sqrtf(var + 1e-5f);
  float w = aw[c], bia = ab[c];
  float* op = xn + (size_t)b * SEQ * CH + c;
#pragma unroll
  for (int q = 0; q < 16; ++q)
    op[(size_t)(lane + q * 32) * CH] = (v[q] - mean) * inv * w + bia;
}

// Gc build: thread = kappa = i*32+m; backward recurrence over s.
// Writes TRANSPOSED GcT[s][kappa] -> fully coalesced stores.
__global__ void film_gc(const float* __restrict__ phi, const float* __restrict__ weR,
                        const float* __restrict__ weI, const float* __restrict__ rR,
                        const float* __restrict__ rI, float* __restrict__ GcR,
                        float* __restrict__ GcI, int L) {
  int kap = blockIdx.x * blockDim.x + threadIdx.x;  // 0..8191
  int i = kap >> 5, m = kap & 31;
  float rr = rR[m], ri = rI[m];
  float TR = 0.f, TI = 0.f;
  for (int s = L - 1; s >= 0; --s) {
    int d = L - 1 - s;
    float pv = phi[d * NH + i];
    TR += weR[d * NMODES + m] * pv;
    TI += weI[d * NMODES + m] * pv;
    float cR = weR[s * NMODES + m], cI = weI[s * NMODES + m];
    float uR = rr * cR - ri * cI;
    float uI = rr * cI + ri * cR;
    GcR[(size_t)s * KTOT + kap] = uR * TR - uI * TI;
    GcI[(size_t)s * KTOT + kap] = uR * TI + uI * TR;
  }
}

// Split-K WMMA f32 GEMM: Mpart[z][o][s] = sum_{kappa in slice} W^T * GcT
// z<8 -> (wr, GcTR) ; z>=8 -> (wi, GcTI). 128x128 block tile, 8 waves.
// Both LDS panels are [row][kk] with LPAD stride; both fragments are v2f loads.
// Next K-panel is prefetched into registers while current panel is consumed.
__global__ __launch_bounds__(256) void film_gemm(
    const float* __restrict__ wr, const float* __restrict__ wi,
    const float* __restrict__ GTR, const float* __restrict__ GTI,
    float* __restrict__ Mpart, int L) {
  __shared__ float As[128 * LPAD];  // [o][kk]
  __shared__ float Bs[128 * LPAD];  // [s][kk]
  const int z = blockIdx.z, zS = z & 7, matSel = z >> 3;
  const float* W = matSel ? wi : wr;
  const float* G = matSel ? GTI : GTR;
  const int oBlock = blockIdx.x * 128;
  const int sBlock = blockIdx.y * 128;
  const int tid = threadIdx.x;
  const int lane = tid & 31, waveId = tid >> 5;
  const int lrow = lane & 15, lhi = lane >> 4;
  const int oW = (waveId & 3) * 32;   // wave o-offset in tile
  const int sW = (waveId >> 2) * 64;  // wave s-offset in tile

  // per-thread fill coordinates (identical for A and B panels: 128 rows x 32)
  int fr[4], fk[4];
#pragma unroll
  for (int q = 0; q < 4; ++q) {
    int flat = (tid + q * 256) * 4;
    fr[q] = flat >> 5;   // panel row (o or s)
    fk[q] = flat & 31;   // kk
  }

  const v8f vzero = {0.f, 0.f, 0.f, 0.f, 0.f, 0.f, 0.f, 0.f};
  v8f acc[2][4];
#pragma unroll
  for (int ot = 0; ot < 2; ++ot)
#pragma unroll
    for (int st = 0; st < 4; ++st) acc[ot][st] = vzero;

  float4 pa[4], pb[4];
  // prologue: load K-step 0 into registers
  {
    const float* Ab = W + (size_t)(zS * 32) * KTOT + (size_t)oBlock * 32;
#pragma unroll
    for (int q = 0; q < 4; ++q) {
      pa[q] = *(const float4*)(Ab + (size_t)(tid + q * 256) * 4);
      pb[q] = *(const float4*)(G + (size_t)(sBlock + fr[q]) * KTOT +
                               (size_t)zS * KSLICE + fk[q]);
    }
  }
#pragma unroll
  for (int q = 0; q < 4; ++q) {
    *(float4*)&As[fr[q] * LPAD + fk[q]] = pa[q];
    *(float4*)&Bs[fr[q] * LPAD + fk[q]] = pb[q];
  }
  __syncthreads();

  for (int kStep = 0; kStep < NSTEPS; ++kStep) {
    // prefetch next panel into registers (global latency overlaps compute)
    if (kStep + 1 < NSTEPS) {
      const float* Ab = W + (size_t)(zS * 32 + kStep + 1) * KTOT + (size_t)oBlock * 32;
#pragma unroll
      for (int q = 0; q < 4; ++q) {
        pa[q] = *(const float4*)(Ab + (size_t)(tid + q * 256) * 4);
        pb[q] = *(const float4*)(G + (size_t)(sBlock + fr[q]) * KTOT +
                                 (size_t)zS * KSLICE + (size_t)(kStep + 1) * 32 + fk[q]);
      }
    }
    // compute: 8 kk-substeps x 8 WMMA
#pragma unroll
    for (int kk = 0; kk < 32; kk += 4) {
      v2f a[2], b[4];
#pragma unroll
      for (int ot = 0; ot < 2; ++ot)
        a[ot] = *(const v2f*)&As[(oW + ot * 16 + lrow) * LPAD + kk + 2 * lhi];
#pragma unroll
      for (int st = 0; st < 4; ++st)
        b[st] = *(const v2f*)&Bs[(sW + st * 16 + lrow) * LPAD + kk + 2 * lhi];
#pragma unroll
      for (int ot = 0; ot < 2; ++ot)
#pragma unroll
        for (int st = 0; st < 4; ++st)
          acc[ot][st] = __builtin_amdgcn_wmma_f32_16x16x4_f32(
              false, a[ot], false, b[st], (short)0, acc[ot][st], false, false);
    }
    __syncthreads();
    if (kStep + 1 < NSTEPS) {
#pragma unroll
      for (int q = 0; q < 4; ++q) {
        *(float4*)&As[fr[q] * LPAD + fk[q]] = pa[q];
        *(float4*)&Bs[fr[q] * LPAD + fk[q]] = pb[q];
      }
    }
    __syncthreads();
  }

  // store: C layout -> o_local = v + 8*lhi, s_local = lrow
  float* dst = Mpart + (size_t)z * NH * L;
#pragma unroll
  for (int ot = 0; ot < 2; ++ot)
#pragma unroll
    for (int st = 0; st < 4; ++st)
#pragma unroll
      for (int v = 0; v < 8; ++v) {
        int o = oBlock + oW + ot * 16 + lhi * 8 + v;
        int s = sBlock + sW + st * 16 + lrow;
        dst[(size_t)o * L + s] = acc[ot][st][v];
      }
}

// Deterministic signed split-K reduction.
__global__ void film_reduce(const float* __restrict__ Mpart, float* __restrict__ M, int L) {
  int id = blockIdx.x * blockDim.x + threadIdx.x;
  int n = NH * L;
  if (id >= n) return;
  float a = 0.f;
#pragma unroll
  for (int zz = 0; zz < 8; ++zz) a += Mpart[(size_t)zz * n + id];
#pragma unroll
  for (int zz = 8; zz < 16; ++zz) a -= Mpart[(size_t)zz * n + id];
  M[id] = a;
}

__global__ void film_zeroQ(float* __restrict__ Q) {
  Q[blockIdx.x * blockDim.x + threadIdx.x] = 0.f;
}

// Q[p, 512-L+s] += mlp_w[k] * sum_o E[p,o] * M[o,s]
__global__ void film_proj(const float* __restrict__ E, const float* __restrict__ M,
                          const float* __restrict__ mlp_w, float* __restrict__ Q,
                          int L, int k) {
  int id = blockIdx.x * blockDim.x + threadIdx.x;
  if (id >= PRED * L) return;
  int p = id / L, s = id % L;
  float acc = 0.f;
  for (int o = 0; o < NH; ++o) acc += E[p * NH + o] * M[(size_t)o * L + s];
  Q[p * SEQ + (SEQ - L) + s] += mlp_w[k] * acc;
}

// out[b,p,j] = mlp_b + sum_t Q[p,t] * xn[b,t,j]
__global__ void film_final(const float* __restrict__ Q, const float* __restrict__ xn,
                           const float* __restrict__ mlp_b, float* __restrict__ out) {
  int id = blockIdx.x * blockDim.x + threadIdx.x;  // b*16384 + p*128 + j
  int b = id >> 14, p = (id >> 7) & 127, j = id & 127;
  float acc = mlp_b[0];
  for (int t = 0; t < SEQ; ++t)
    acc += Q[p * SEQ + t] * xn[(size_t)b * SEQ * CH + (size_t)t * CH + j];
  out[id] = acc;
}

// ------------------------------- launcher ----------------------------------
extern "C" void kernel_launch(void* const* d_in, const int* in_sizes, int n_in,
                              void* d_out, int out_size, void* d_ws, size_t ws_size,
                              hipStream_t stream) {
  (void)in_sizes; (void)n_in; (void)out_size; (void)ws_size;
  const float* x_enc  = (const float*)d_in[0];
  const float* aw     = (const float*)d_in[1];
  const float* ab     = (const float*)d_in[2];
  const float* w_real = (const float*)d_in[3];
  const float* w_imag = (const float*)d_in[4];
  const float* mlp_w  = (const float*)d_in[5];
  const float* mlp_b  = (const float*)d_in[6];
  float* out = (float*)d_out;

  build_constants();  // deterministic; rewrites g_hc identically each call

  char* ws = (char*)d_ws;
  size_t off = 0;
  auto take = [&](size_t bytes) { size_t o = off; off += (bytes + 255) & ~(size_t)255; return o; };
  const size_t CONST_OFF = take(sizeof(HostConsts));
  const size_t XN_OFF    = take((size_t)NBATCH * SEQ * CH * 4);
  const size_t GCR_OFF   = take((size_t)KTOT * 512 * 4);
  const size_t GCI_OFF   = take((size_t)KTOT * 512 * 4);
  const size_t MPART_OFF = take((size_t)16 * NH * 512 * 4);
  const size_t M_OFF     = take((size_t)NH * 512 * 4);
  const size_t Q_OFF     = take((size_t)PRED * SEQ * 4);
  // total workspace ~46 MB

  hipMemcpyAsync(ws + CONST_OFF, &g_hc, sizeof(HostConsts),
                 hipMemcpyHostToDevice, stream);

  float* c_phi = (float*)(ws + CONST_OFF + offsetof(HostConsts, phi));
  float* c_weR = (float*)(ws + CONST_OFF + offsetof(HostConsts, weR));
  float* c_weI = (float*)(ws + CONST_OFF + offsetof(HostConsts, weI));
  float* c_rR  = (float*)(ws + CONST_OFF + offsetof(HostConsts, rR));
  float* c_rI  = (float*)(ws + CONST_OFF + offsetof(HostConsts, rI));
  float* c_E   = (float*)(ws + CONST_OFF + offsetof(HostConsts, E));
  float* xn    = (float*)(ws + XN_OFF);
  float* GcR   = (float*)(ws + GCR_OFF);
  float* GcI   = (float*)(ws + GCI_OFF);
  float* Mpart = (float*)(ws + MPART_OFF);
  float* M     = (float*)(ws + M_OFF);
  float* Q     = (float*)(ws + Q_OFF);

  film_revin<<<64, 256, 0, stream>>>(x_enc, aw, ab, xn);
  film_zeroQ<<<(PRED * SEQ) / 256, 256, 0, stream>>>(Q);

  int phiOff = 0, weOff = 0;
  for (int k = 0; k < 3; ++k) {
    const int L = PRED << k;
    film_gc<<<KTOT / 256, 256, 0, stream>>>(c_phi + phiOff, c_weR + weOff,
                                            c_weI + weOff, c_rR + k * NMODES,
                                            c_rI + k * NMODES, GcR, GcI, L);
    film_gemm<<<dim3(2, L / 128, 16), 256, 0, stream>>>(
        w_real + (size_t)k * NH * KTOT, w_imag + (size_t)k * NH * KTOT,
        GcR, GcI, Mpart, L);
    film_reduce<<<(NH * L) / 256, 256, 0, stream>>>(Mpart, M, L);
    film_proj<<<(PRED * L) / 256, 256, 0, stream>>>(c_E + k * PRED * NH, M,
                                                    mlp_w, Q, L, k);
    phiOff += L * NH;
    weOff  += L * NMODES;
  }
  film_final<<<(NBATCH * PRED * CH) / 256, 256, 0, stream>>>(Q, xn, mlp_b, out);
}